// CTC_layer_69260642615580
// MI455X (gfx1250) — compile-verified
//
#include <hip/hip_runtime.h>
#include <stdint.h>

// Problem sizes from the reference (fixed).
#define B_   32
#define T_   1024
#define V_   1024
#define L_   100
#define S_   201            // 2L+1 extended states
#define BLANK_ (V_ - 1)
#define NEGV (-1e30f)

// ---- CDNA5 TDM availability (amdgpu-toolchain / clang-23 6-arg form) ----
#ifndef USE_TDM
#if defined(__has_builtin)
#if __has_builtin(__builtin_amdgcn_tensor_load_to_lds) && \
    __has_builtin(__builtin_amdgcn_s_wait_tensorcnt)
#define USE_TDM 1
#endif
#endif
#endif
#ifndef USE_TDM
#define USE_TDM 0
#endif

typedef unsigned int u32x4 __attribute__((ext_vector_type(4)));
typedef int          i32x8 __attribute__((ext_vector_type(8)));
typedef int          i32x4 __attribute__((ext_vector_type(4)));

constexpr int ROWS_PER_BLK = 8;                    // 8 rows x 4KB = 32KB LDS tile
constexpr int TILE_ELEMS   = ROWS_PER_BLK * V_;    // 8192 f32

// ---------------------------------------------------------------------------
// Kernel A: per-row logsumexp of logits. One 32KB TDM tile -> LDS per block,
// one wave32 per row; row is pulled from LDS once as 8x ds_load_b128 into
// registers, then max + exp-sum both run from registers.
// Streams the full 128MB once (-> resident in 192MB L2 for kernel B).
// ---------------------------------------------------------------------------
__global__ __launch_bounds__(256) void lse_kernel(const float* __restrict__ logits,
                                                  float* __restrict__ lse)
{
    __shared__ float tile[TILE_ELEMS];
    const int group = blockIdx.x;  // covers rows [group*8, group*8+8)
    const float* src = logits + (size_t)group * TILE_ELEMS;

#if USE_TDM
    if (threadIdx.x < 32u) {  // one wave issues the DMA (TDM ignores EXEC)
        const unsigned long long ga = (unsigned long long)(uintptr_t)src;
        const unsigned int lds_addr = (unsigned int)(uintptr_t)&tile[0]; // low32 of flat = LDS offset

        u32x4 g0;
        g0[0] = 1u;                                          // count=1, user descriptor
        g0[1] = lds_addr;                                    // lds_addr [63:32]
        g0[2] = (unsigned int)ga;                            // global_addr[31:0]
        g0[3] = ((unsigned int)(ga >> 32) & 0x01FFFFFFu)     // global_addr[56:32]
                | (2u << 30);                                // type=2 ("image")
        i32x8 g1;
        g1[0] = (int)(2u << 16);                             // data_size=4B; no multicast/pad
        g1[1] = (int)((unsigned)TILE_ELEMS << 16);           // tensor_dim0[15:0] -> [31:16]
        g1[2] = (int)(((unsigned)TILE_ELEMS >> 16)           // tensor_dim0[31:16]
                | (1u << 16));                               // tensor_dim1 = 1
        g1[3] = (int)((unsigned)TILE_ELEMS << 16);           // tile_dim0 = 8192 (1D, contiguous)
        g1[4] = 0;                                           // tile_dim1/2 unused
        g1[5] = (int)TILE_ELEMS;                             // tensor_dim0_stride
        g1[6] = 0;
        g1[7] = 0;
        i32x4 g2 = {0, 0, 0, 0};
        i32x4 g3 = {0, 0, 0, 0};
        i32x8 g4 = {0, 0, 0, 0, 0, 0, 0, 0};                 // extra group (6-arg form)
        __builtin_amdgcn_tensor_load_to_lds(g0, g1, g2, g3, g4, 0);
        __builtin_amdgcn_s_wait_tensorcnt(0);
        asm volatile("" ::: "memory");
    }
#else
    for (int i = threadIdx.x; i < TILE_ELEMS; i += 256) tile[i] = src[i];
#endif
    __syncthreads();

    const int wave = threadIdx.x >> 5;
    const int lane = threadIdx.x & 31;
    const float4* row4 = reinterpret_cast<const float4*>(&tile[wave * V_]);

    float4 v[8];                               // 1024 f32 / 32 lanes = 8 x float4
#pragma unroll
    for (int i = 0; i < 8; ++i) v[i] = row4[i * 32 + lane];

    float m = NEGV;
#pragma unroll
    for (int i = 0; i < 8; ++i) {
        m = fmaxf(m, fmaxf(fmaxf(v[i].x, v[i].y), fmaxf(v[i].z, v[i].w)));
    }
#pragma unroll
    for (int off = 16; off >= 1; off >>= 1) m = fmaxf(m, __shfl_xor(m, off, 32));

    float s = 0.0f;
#pragma unroll
    for (int i = 0; i < 8; ++i) {
        s += __expf(v[i].x - m) + __expf(v[i].y - m) +
             __expf(v[i].z - m) + __expf(v[i].w - m);
    }
#pragma unroll
    for (int off = 16; off >= 1; off >>= 1) s += __shfl_xor(s, off, 32);

    if (lane == 0) lse[group * ROWS_PER_BLK + wave] = m + __logf(s);
}

// logaddexp of three terms (NEG sentinels underflow to 0 in expf, no NaN)
__device__ __forceinline__ float lae3(float a, float b, float c)
{
    float m = fmaxf(a, fmaxf(b, c));
    return m + __logf(__expf(a - m) + __expf(b - m) + __expf(c - m));
}

// ---------------------------------------------------------------------------
// Kernel B: CTC forward recursion. One wave32 per batch element; lane l owns
// states 7l..7l+6 (blocked) -> only 2 cross-lane shuffles per time step.
// Label-prob gathers (7/lane/step) hit L2 (logits fully L2-resident) and are
// software-pipelined DEPTH=8 steps ahead. The pipeline ring is indexed only
// by compile-time constants (outer loop strides by DEPTH, inner loop fully
// unrolled) so buf[][] is guaranteed register-resident — no scratch on the
// critical recursion path.
// ---------------------------------------------------------------------------
__global__ __launch_bounds__(32) void ctc_forward(const int* __restrict__ targets,
                                                  const float* __restrict__ logits,
                                                  const float* __restrict__ lse,
                                                  float* __restrict__ loss)
{
    const int b    = blockIdx.x;
    const int lane = threadIdx.x;
    const int*   tg = targets + b * L_;
    const float* lg = logits + (size_t)b * T_ * V_;
    const float* ls = lse + b * T_;

    int  e[7];
    bool skip[7];
#pragma unroll
    for (int r = 0; r < 7; ++r) {
        const int s = lane * 7 + r;
        int  lab = BLANK_;
        bool sk  = false;
        if ((s & 1) && s < S_) {              // odd state -> real label
            const int li = (s - 1) >> 1;
            lab = tg[li];
            if (s >= 3) sk = (lab != tg[li - 1]);  // skip iff label differs
        }
        e[r] = lab;
        skip[r] = sk;
    }

    // t = 0 init: alpha[0] = lp(blank), alpha[1] = lp(label1), rest NEG
    float alpha[7];
#pragma unroll
    for (int r = 0; r < 7; ++r) alpha[r] = NEGV;
    {
        const float lse0 = ls[0];
        if (lane == 0) {
            alpha[0] = lg[e[0]] - lse0;
            alpha[1] = lg[e[1]] - lse0;
        }
    }

    constexpr int DEPTH = 8;                  // prefetch pipeline depth
    float buf[DEPTH][7];
    float lsebuf[DEPTH];
#pragma unroll
    for (int d = 0; d < DEPTH; ++d) {         // prefill t = 1 .. DEPTH
        const int t = 1 + d;
        const float* p = lg + (size_t)t * V_;
#pragma unroll
        for (int r = 0; r < 7; ++r) buf[d][r] = p[e[r]];
        lsebuf[d] = ls[t];
    }

    for (int t0 = 1; t0 < T_; t0 += DEPTH) {
#pragma unroll
        for (int k = 0; k < DEPTH; ++k) {     // buf index k is compile-time
            const int t = t0 + k;
            if (t >= T_) break;               // uniform tail guard

            const float l0 = lsebuf[k];
            float lp[7];
#pragma unroll
            for (int r = 0; r < 7; ++r) lp[r] = buf[k][r] - l0;

            const int tp = t + DEPTH;         // refill this pipeline slot
            if (tp < T_) {
                const float* p = lg + (size_t)tp * V_;
#pragma unroll
                for (int r = 0; r < 7; ++r) buf[k][r] = p[e[r]];
                lsebuf[k] = ls[tp];
            }

            // neighbor lane's top two states: s = 7l-1 and 7l-2
            float top1 = __shfl_up(alpha[6], 1, 32);
            float top2 = __shfl_up(alpha[5], 1, 32);
            if (lane == 0) { top1 = NEGV; top2 = NEGV; }

            float na[7];
            na[0] = lae3(alpha[0], top1,     skip[0] ? top2     : NEGV) + lp[0];
            na[1] = lae3(alpha[1], alpha[0], skip[1] ? top1     : NEGV) + lp[1];
#pragma unroll
            for (int r = 2; r < 7; ++r)
                na[r] = lae3(alpha[r], alpha[r - 1], skip[r] ? alpha[r - 2] : NEGV) + lp[r];
#pragma unroll
            for (int r = 0; r < 7; ++r) alpha[r] = na[r];
        }
    }

    // loss_b = -logaddexp(alpha[S-1], alpha[S-2]); s=200 -> lane28 r4, s=199 -> lane28 r3
    const float aLast  = __shfl(alpha[4], 28, 32);
    const float aLast2 = __shfl(alpha[3], 28, 32);
    if (lane == 0) {
        const float m = fmaxf(aLast, aLast2);
        loss[b] = -(m + __logf(__expf(aLast - m) + __expf(aLast2 - m)));
    }
}

// ---------------------------------------------------------------------------
// Kernel C: deterministic mean of the 32 per-batch losses (shuffle tree).
// ---------------------------------------------------------------------------
__global__ __launch_bounds__(32) void mean_kernel(const float* __restrict__ loss,
                                                  float* __restrict__ out)
{
    const int lane = threadIdx.x;
    float v = loss[lane];  // B_ == 32 == wave32
#pragma unroll
    for (int off = 16; off >= 1; off >>= 1) v += __shfl_xor(v, off, 32);
    if (lane == 0) out[0] = v * (1.0f / (float)B_);
}

extern "C" void kernel_launch(void* const* d_in, const int* in_sizes, int n_in,
                              void* d_out, int out_size, void* d_ws, size_t ws_size,
                              hipStream_t stream)
{
    (void)in_sizes; (void)n_in; (void)out_size; (void)ws_size;
    const int*   targets = (const int*)d_in[0];   // [B, L] int32
    const float* logits  = (const float*)d_in[1]; // [B, T, V] f32

    float* lse  = (float*)d_ws;                   // B*T floats (128 KB)
    float* loss = lse + (size_t)B_ * T_;          // B floats

    lse_kernel<<<(B_ * T_) / ROWS_PER_BLK, 256, 0, stream>>>(logits, lse);
    ctc_forward<<<B_, 32, 0, stream>>>(targets, logits, lse, loss);
    mean_kernel<<<1, 32, 0, stream>>>(loss, (float*)d_out);
}